// RecurrentGNN_27298812134108
// MI455X (gfx1250) — compile-verified
//
#include <hip/hip_runtime.h>
#include <hip/hip_bf16.h>

typedef __attribute__((ext_vector_type(2))) float v2f;
typedef __attribute__((ext_vector_type(8))) float v8f;

__device__ __forceinline__ float sigmoidf_(float v) {
    return 1.0f / (1.0f + __expf(-v));
}

__device__ __forceinline__ void atomic_add_f32(float* p, float v) {
    __hip_atomic_fetch_add(p, v, __ATOMIC_RELAXED, __HIP_MEMORY_SCOPE_AGENT);
}

// ---------------- zero scratch (vectorized, counts are multiples of 4) ----------------
__global__ void zero4_kernel(float4* __restrict__ p, long n4) {
    long i = (long)blockIdx.x * blockDim.x + threadIdx.x;
    if (i < n4) p[i] = make_float4(0.f, 0.f, 0.f, 0.f);
}

// ---------------- weighted degrees ----------------
__global__ void degrees_kernel(const int* __restrict__ row, const int* __restrict__ col,
                               const float* __restrict__ w,
                               float* __restrict__ dout, float* __restrict__ din, int E) {
    int e = blockIdx.x * blockDim.x + threadIdx.x;
    if (e >= E) return;
    float we = w[e];
    atomic_add_f32(&dout[row[e]], we);
    atomic_add_f32(&din[col[e]], we);
}

// in-place reciprocal over 2N contiguous floats (deg_out | deg_in)
__global__ void invert_kernel(float* __restrict__ p, int n2) {
    int i = blockIdx.x * blockDim.x + threadIdx.x;
    if (i < n2) p[i] = 1.0f / p[i];
}

// ---------------- h = sigmoid(x @ W_lin2 + b) ----------------
__global__ void lin2_kernel(const float* __restrict__ x, const float* __restrict__ w,
                            const float* __restrict__ b, float* __restrict__ h, int N) {
    int idx = blockIdx.x * blockDim.x + threadIdx.x;
    if (idx >= N * 64) return;
    int n = idx >> 6, c = idx & 63;
    const float* xr = x + (long)n * 24;
    float acc = b[c];
#pragma unroll
    for (int k = 0; k < 24; ++k) acc = fmaf(xr[k], w[k * 64 + c], acc);
    h[idx] = sigmoidf_(acc);
}

// ---------------- fused 2-direction diffusion propagate (float4 per thread) ----------
// tx_o[col] += h[row] / deg_out[row]
// tx_i[row] += h[col] / deg_in[row]      (norm_in = (1/deg_in)[row] in the reference)
__global__ void prop_kernel(const int* __restrict__ row, const int* __restrict__ col,
                            const float* __restrict__ h,
                            const float* __restrict__ inv_out, const float* __restrict__ inv_in,
                            float* __restrict__ txo, float* __restrict__ txi, int E) {
    long idx = (long)blockIdx.x * blockDim.x + threadIdx.x;
    if (idx >= (long)E * 16) return;
    int e = (int)(idx >> 4);
    int c4 = (int)(idx & 15) << 2;
    int r = row[e], cl = col[e];
    float so = inv_out[r];
    float si = inv_in[r];
    float4 vr = *(const float4*)(h + (long)r * 64 + c4);
    float4 vc = *(const float4*)(h + (long)cl * 64 + c4);
    float* po = txo + (long)cl * 64 + c4;
    float* pi = txi + (long)r * 64 + c4;
    atomic_add_f32(po + 0, vr.x * so);
    atomic_add_f32(po + 1, vr.y * so);
    atomic_add_f32(po + 2, vr.z * so);
    atomic_add_f32(po + 3, vr.w * so);
    atomic_add_f32(pi + 0, vc.x * si);
    atomic_add_f32(pi + 1, vc.y * si);
    atomic_add_f32(pi + 2, vc.z * si);
    atomic_add_f32(pi + 3, vc.w * si);
}

// ---------------- Wsum = w[0,0][:64] + w[1,0][:64] for z and h ----------------
__global__ void wsum_kernel(const float* __restrict__ wz, const float* __restrict__ wh,
                            float* __restrict__ wsz, float* __restrict__ wsh) {
    int i = blockIdx.x * blockDim.x + threadIdx.x;
    if (i >= 2 * 4096) return;
    if (i < 4096) wsz[i] = wz[i] + wz[2 * 8192 + i];        // (k1,k2)=(0,0) + (1,0)
    else { int j = i - 4096; wsh[j] = wh[j] + wh[2 * 8192 + j]; }
}

// ---------------- fused GRU gates (WMMA f32) + output GEMM ----------------
// Block = 256 threads = 8 waves; block owns a 16-node tile.
// wave w: mat = w/4 (0 -> z-gate, 1 -> h~), ntile = w%4 (16-column slab).
// Each wave accumulates 3 GEMM phases (h*Wsum, txo*W01, txi*W11) with K=64
// via v_wmma_f32_16x16x4_f32 (16 K-steps per phase).
__global__ __launch_bounds__(256) void gru_out_kernel(
    const float* __restrict__ h, const float* __restrict__ txo, const float* __restrict__ txi,
    const float* __restrict__ wsum_z, const float* __restrict__ wsum_h,
    const float* __restrict__ wz, const float* __restrict__ wh,
    const float* __restrict__ bz, const float* __restrict__ bh,
    const float* __restrict__ wout, const float* __restrict__ bout,
    float* __restrict__ out, int N) {
    __shared__ float A[3][16][68];   // h / txo / txi tiles; pad 68 -> 16B-aligned rows,
                                     // bank stride 4/row => conflict-free ds_load_b64
    __shared__ float pre[16][130];   // cols 0..63 = z_pre, 64..127 = ht_pre
    __shared__ float s[16][66];      // sigmoid(h_new)

    const int n0 = blockIdx.x * 16;
    const int tid = threadIdx.x;

    // cooperative stage of the three 16x64 A-tiles, 16B granules (exactly 1 per thread)
    {
        int m = tid >> 4, c = (tid & 15) << 2;
        int nr = n0 + m; if (nr >= N) nr = N - 1;
        long g = (long)nr * 64 + c;
        *(float4*)&A[0][m][c] = *(const float4*)(h + g);
        *(float4*)&A[1][m][c] = *(const float4*)(txo + g);
        *(float4*)&A[2][m][c] = *(const float4*)(txi + g);
    }
    __syncthreads();

    const int wave = tid >> 5;
    const int lane = tid & 31;
    const int mat = wave >> 2;             // 0 = z, 1 = h~
    const int nt = wave & 3;               // column tile
    const int mrow = lane & 15;            // A row / B column (mod 16)
    const int khalf = (lane < 16) ? 0 : 2; // K sub-pair per lane half
    const int ncol = nt * 16 + mrow;       // output column this lane owns

    // per-lane B base pointers (pre-offset by ncol -> all loads use immediate offsets)
    const float* B0 = (mat ? wsum_h : wsum_z) + ncol;          // combined k=0 term
    const float* B1 = (mat ? wh : wz) + 128 * 64 + ncol;       // w[0,1][:64]
    const float* B2 = (mat ? wh : wz) + 3 * 128 * 64 + ncol;   // w[1,1][:64]

    v8f acc = {};
#pragma unroll
    for (int kk = 0; kk < 16; ++kk) {
        int kb = kk * 4 + khalf;
        v2f a, b;
        a.x = A[0][mrow][kb]; a.y = A[0][mrow][kb + 1];
        b.x = B0[kb * 64];    b.y = B0[(kb + 1) * 64];
        acc = __builtin_amdgcn_wmma_f32_16x16x4_f32(false, a, false, b, (short)0, acc, false, false);
    }
#pragma unroll
    for (int kk = 0; kk < 16; ++kk) {
        int kb = kk * 4 + khalf;
        v2f a, b;
        a.x = A[1][mrow][kb]; a.y = A[1][mrow][kb + 1];
        b.x = B1[kb * 64];    b.y = B1[(kb + 1) * 64];
        acc = __builtin_amdgcn_wmma_f32_16x16x4_f32(false, a, false, b, (short)0, acc, false, false);
    }
#pragma unroll
    for (int kk = 0; kk < 16; ++kk) {
        int kb = kk * 4 + khalf;
        v2f a, b;
        a.x = A[2][mrow][kb]; a.y = A[2][mrow][kb + 1];
        b.x = B2[kb * 64];    b.y = B2[(kb + 1) * 64];
        acc = __builtin_amdgcn_wmma_f32_16x16x4_f32(false, a, false, b, (short)0, acc, false, false);
    }

    const float bias = (mat ? bh : bz)[ncol];
#pragma unroll
    for (int r = 0; r < 8; ++r) {
        int mr = (lane < 16) ? r : (r + 8);  // C layout: VGPR r -> M=r | M=r+8
        pre[mr][mat * 64 + ncol] = acc[r] + bias;
    }
    __syncthreads();

    // gates: z = sigma(zp); ht = tanh(hp); h_new = (1-z)*ht; s = sigma(h_new)
    for (int i = tid; i < 16 * 64; i += 256) {
        int m = i >> 6, n = i & 63;
        float z = sigmoidf_(pre[m][n]);
        float ht = tanhf(pre[m][64 + n]);
        s[m][n] = sigmoidf_((1.0f - z) * ht);
    }
    __syncthreads();

    // out[16,12] = s[16,64] @ wout[64,12] + bout
    if (tid < 16 * 12) {
        int m = tid / 12, o = tid % 12;
        if (n0 + m < N) {
            float a = bout[o];
#pragma unroll
            for (int k = 0; k < 64; ++k) a = fmaf(s[m][k], wout[k * 12 + o], a);
            out[(long)(n0 + m) * 12 + o] = a;
        }
    }
}

extern "C" void kernel_launch(void* const* d_in, const int* in_sizes, int n_in,
                              void* d_out, int out_size, void* d_ws, size_t ws_size,
                              hipStream_t stream) {
    const float* x      = (const float*)d_in[0];
    const int*   ei     = (const int*)d_in[1];
    const float* ew     = (const float*)d_in[2];
    const float* w_lin2 = (const float*)d_in[3];
    const float* b_lin2 = (const float*)d_in[4];
    const float* w_z    = (const float*)d_in[5];
    const float* b_z    = (const float*)d_in[6];
    // d_in[7], d_in[8] = w_r, b_r : dead code (h0 == 0 makes the r-gate unused)
    const float* w_h    = (const float*)d_in[9];
    const float* b_h    = (const float*)d_in[10];
    const float* w_out  = (const float*)d_in[11];
    const float* b_out  = (const float*)d_in[12];
    float* out = (float*)d_out;

    const int N = in_sizes[0] / 24;
    const int E = in_sizes[2];
    const int* row = ei;
    const int* col = ei + E;

    float* ws = (float*)d_ws;
    float* deg    = ws;                    // deg_out[N] | deg_in[N]
    float* deg_in = ws + N;
    float* hbuf   = ws + 2L * N;           // N*64
    float* txo    = hbuf + 64L * N;        // N*64
    float* txi    = txo + 64L * N;         // N*64
    float* wsz    = txi + 64L * N;         // 4096
    float* wsh    = wsz + 4096;            // 4096
    (void)ws_size; (void)n_in; (void)out_size;

    const int B = 256;
    // zero degrees and tx accumulators (must be re-zeroed every call)
    long zn1 = (2L * N) / 4;               // N is a multiple of 2 -> 2N % 4 == 0
    zero4_kernel<<<(int)((zn1 + B - 1) / B), B, 0, stream>>>((float4*)deg, zn1);
    long zn2 = (128L * N) / 4;
    zero4_kernel<<<(int)((zn2 + B - 1) / B), B, 0, stream>>>((float4*)txo, zn2);

    degrees_kernel<<<(E + B - 1) / B, B, 0, stream>>>(row, col, ew, deg, deg_in, E);
    invert_kernel<<<(2 * N + B - 1) / B, B, 0, stream>>>(deg, 2 * N);

    lin2_kernel<<<(N * 64 + B - 1) / B, B, 0, stream>>>(x, w_lin2, b_lin2, hbuf, N);

    wsum_kernel<<<(8192 + B - 1) / B, B, 0, stream>>>(w_z, w_h, wsz, wsh);

    long pt = (long)E * 16;
    prop_kernel<<<(int)((pt + B - 1) / B), B, 0, stream>>>(row, col, hbuf, deg, deg_in,
                                                           txo, txi, E);

    gru_out_kernel<<<(N + 15) / 16, 256, 0, stream>>>(hbuf, txo, txi, wsz, wsh,
                                                      w_z, w_h, b_z, b_h,
                                                      w_out, b_out, out, N);
}